// CrossAttention_10204842295864
// MI455X (gfx1250) — compile-verified
//
#include <hip/hip_runtime.h>

// out_b = X_b @ [ Wq^T Wk (C_b^T C_b) Wv^T ] * 128^-0.5    (associativity rewrite,
// no softmax in the reference -> O(S*D^2) instead of O(S^2*D))

typedef float v2f __attribute__((ext_vector_type(2)));
typedef float v8f __attribute__((ext_vector_type(8)));

constexpr int Dm    = 128;   // feature dim
constexpr int SQ    = 4096;
constexpr int SKV   = 4096;
constexpr int NB    = 4;     // batch
constexpr int SPLIT = 16;    // K-split for Gram kernel
constexpr int ROWS  = SKV / SPLIT;  // 256 kv rows per workgroup
#define ATT_SCALE 0.08838834764831845f  // 128^-0.5

// D = A(16x4,f32) * B(4x16,f32) + C(16x16,f32)
static __device__ __forceinline__ v8f wmma4(v2f a, v2f b, v8f c) {
  return __builtin_amdgcn_wmma_f32_16x16x4_f32(false, a, false, b, (short)0, c,
                                               false, false);
}

// ---------------------------------------------------------------------------
// Kernel 1: per-(batch, split) partial Gram  P[b][sp] = C_chunk^T @ C_chunk
// ---------------------------------------------------------------------------
__global__ __launch_bounds__(256) void gram_partial_kernel(
    const float* __restrict__ Ctx, float* __restrict__ P) {
  __shared__ float Cs[128 * Dm];  // 64 KB staging of a 128-row chunk of context
  const int b    = blockIdx.x / SPLIT;
  const int sp   = blockIdx.x % SPLIT;
  const int tid  = threadIdx.x;
  const int wave = tid >> 5, lane = tid & 31;
  const int half = lane >> 4, mn = lane & 15;
  const int k0   = half * 2;          // K offset pair per ISA A/B f32 layout
  const int mbase = wave * 16;        // each wave owns a 16-row output strip
  const float* Cb = Ctx + ((size_t)b * SKV + (size_t)sp * ROWS) * Dm;

  v8f acc[8] = {};
  for (int c = 0; c < ROWS; c += 128) {
    __syncthreads();
    const float4* src = (const float4*)(Cb + (size_t)c * Dm);
    float4* dst = (float4*)Cs;
    for (int i = tid; i < 128 * Dm / 4; i += 256) dst[i] = src[i];
    __syncthreads();
#pragma unroll 4
    for (int k = 0; k < 128; k += 4) {
      v2f a;  // A = C^T tile: A[m][kk] = C[kk][m]
      a.x = Cs[(k + k0) * Dm + mbase + mn];
      a.y = Cs[(k + k0 + 1) * Dm + mbase + mn];
#pragma unroll
      for (int t = 0; t < 8; ++t) {
        v2f bm;  // B = C tile: B[kk][n]
        bm.x = Cs[(k + k0) * Dm + t * 16 + mn];
        bm.y = Cs[(k + k0 + 1) * Dm + t * 16 + mn];
        acc[t] = wmma4(a, bm, acc[t]);
      }
    }
  }
  float* Pp = P + ((size_t)b * SPLIT + sp) * Dm * Dm;
#pragma unroll
  for (int t = 0; t < 8; ++t)
#pragma unroll
    for (int i = 0; i < 8; ++i)
      Pp[(mbase + i + 8 * half) * Dm + t * 16 + mn] = acc[t][i];
}

// ---------------------------------------------------------------------------
// Kernel 2: T_b = Wq^T @ ( Wk @ ( (sum_sp P[b][sp]) @ Wv^T ) ) * scale
// one workgroup per batch; fixed-order split reduction => deterministic
// ---------------------------------------------------------------------------
__global__ __launch_bounds__(256) void sandwich_kernel(
    const float* __restrict__ P, const float* __restrict__ Wq,
    const float* __restrict__ Wk, const float* __restrict__ Wv,
    float* __restrict__ U, float* __restrict__ R, float* __restrict__ T) {
  const int b    = blockIdx.x;
  const int tid  = threadIdx.x;
  const int wave = tid >> 5, lane = tid & 31;
  const int half = lane >> 4, mn = lane & 15;
  const int k0   = half * 2;
  const int mbase = wave * 16;
  const float* Pb = P + (size_t)b * SPLIT * Dm * Dm;
  float* Ub = U + b * Dm * Dm;
  float* Rb = R + b * Dm * Dm;
  float* Tb = T + b * Dm * Dm;

  {  // Stage 1: U = G @ Wv^T   (G[m][k] summed over splits on the fly)
    v8f acc[8] = {};
    for (int k = 0; k < Dm; k += 4) {
      float ax = 0.f, ay = 0.f;
      const int ro = (mbase + mn) * Dm + k + k0;
#pragma unroll
      for (int sp = 0; sp < SPLIT; ++sp) {
        ax += Pb[sp * Dm * Dm + ro];
        ay += Pb[sp * Dm * Dm + ro + 1];
      }
      v2f a; a.x = ax; a.y = ay;
#pragma unroll
      for (int t = 0; t < 8; ++t) {
        v2f bm;  // B[kk][n] = Wv[n][kk]
        bm.x = Wv[(t * 16 + mn) * Dm + k + k0];
        bm.y = Wv[(t * 16 + mn) * Dm + k + k0 + 1];
        acc[t] = wmma4(a, bm, acc[t]);
      }
    }
#pragma unroll
    for (int t = 0; t < 8; ++t)
#pragma unroll
      for (int i = 0; i < 8; ++i)
        Ub[(mbase + i + 8 * half) * Dm + t * 16 + mn] = acc[t][i];
  }
  __threadfence(); __syncthreads();
  {  // Stage 2: R = Wk @ U
    v8f acc[8] = {};
    for (int k = 0; k < Dm; k += 4) {
      v2f a;
      a.x = Wk[(mbase + mn) * Dm + k + k0];
      a.y = Wk[(mbase + mn) * Dm + k + k0 + 1];
#pragma unroll
      for (int t = 0; t < 8; ++t) {
        v2f bm;
        bm.x = Ub[(k + k0) * Dm + t * 16 + mn];
        bm.y = Ub[(k + k0 + 1) * Dm + t * 16 + mn];
        acc[t] = wmma4(a, bm, acc[t]);
      }
    }
#pragma unroll
    for (int t = 0; t < 8; ++t)
#pragma unroll
      for (int i = 0; i < 8; ++i)
        Rb[(mbase + i + 8 * half) * Dm + t * 16 + mn] = acc[t][i];
  }
  __threadfence(); __syncthreads();
  {  // Stage 3: T = (Wq^T @ R) * scale   (A[m][kk] = Wq[kk][m])
    v8f acc[8] = {};
    for (int k = 0; k < Dm; k += 4) {
      v2f a;
      a.x = Wq[(k + k0) * Dm + mbase + mn];
      a.y = Wq[(k + k0 + 1) * Dm + mbase + mn];
#pragma unroll
      for (int t = 0; t < 8; ++t) {
        v2f bm;
        bm.x = Rb[(k + k0) * Dm + t * 16 + mn];
        bm.y = Rb[(k + k0 + 1) * Dm + t * 16 + mn];
        acc[t] = wmma4(a, bm, acc[t]);
      }
    }
#pragma unroll
    for (int t = 0; t < 8; ++t)
#pragma unroll
      for (int i = 0; i < 8; ++i)
        Tb[(mbase + i + 8 * half) * Dm + t * 16 + mn] = acc[t][i] * ATT_SCALE;
  }
}

// ---------------------------------------------------------------------------
// Kernel 3: out_b = X_b @ T_b   (bandwidth-bound streaming GEMM)
// ---------------------------------------------------------------------------
__global__ __launch_bounds__(256) void out_kernel(const float* __restrict__ X,
                                                  const float* __restrict__ T,
                                                  float* __restrict__ out) {
  __shared__ float Ts[Dm * Dm];  // 64 KB: T_b staged once per workgroup
  constexpr int NRB = SQ / 128;  // 32 row-blocks per batch
  const int b    = blockIdx.x / NRB;
  const int rblk = blockIdx.x % NRB;
  const int tid  = threadIdx.x;
  {
    const float4* src = (const float4*)(T + (size_t)b * Dm * Dm);
    float4* dst = (float4*)Ts;
    for (int i = tid; i < Dm * Dm / 4; i += 256) dst[i] = src[i];
  }
  __syncthreads();

  const int wave = tid >> 5, lane = tid & 31;
  const int half = lane >> 4, mn = lane & 15;
  const int k0   = half * 2;
  const int row  = rblk * 128 + wave * 16 + mn;
  const float* Xrow = X + ((size_t)b * SQ + row) * Dm;

  v8f acc[8] = {};
#pragma unroll 4
  for (int k = 0; k < Dm; k += 4) {
    v2f a = *(const v2f*)(Xrow + k + k0);  // 8B-aligned vector load
#pragma unroll
    for (int t = 0; t < 8; ++t) {
      v2f bm;
      bm.x = Ts[(k + k0) * Dm + t * 16 + mn];
      bm.y = Ts[(k + k0 + 1) * Dm + t * 16 + mn];
      acc[t] = wmma4(a, bm, acc[t]);
    }
  }
  float* ob = out + ((size_t)b * SQ + (size_t)rblk * 128 + wave * 16) * Dm;
#pragma unroll
  for (int t = 0; t < 8; ++t)
#pragma unroll
    for (int i = 0; i < 8; ++i)
      ob[(i + 8 * half) * Dm + t * 16 + mn] = acc[t][i];
}

// ---------------------------------------------------------------------------
extern "C" void kernel_launch(void* const* d_in, const int* in_sizes, int n_in,
                              void* d_out, int out_size, void* d_ws, size_t ws_size,
                              hipStream_t stream) {
  const float* context = (const float*)d_in[0];  // [4,4096,128]
  const float* X       = (const float*)d_in[1];  // [4,4096,128]
  const float* Wq      = (const float*)d_in[2];  // [128,128]
  const float* Wk      = (const float*)d_in[3];
  const float* Wv      = (const float*)d_in[4];
  float* out = (float*)d_out;

  // workspace layout (floats): P | U | R | T   (~4.75 MB total)
  float* P = (float*)d_ws;                       // NB*SPLIT*128*128
  float* U = P + (size_t)NB * SPLIT * Dm * Dm;
  float* R = U + (size_t)NB * Dm * Dm;
  float* T = R + (size_t)NB * Dm * Dm;

  gram_partial_kernel<<<NB * SPLIT, 256, 0, stream>>>(context, P);
  sandwich_kernel<<<NB, 256, 0, stream>>>(P, Wq, Wk, Wv, U, R, T);
  out_kernel<<<NB * (SQ / 128), 256, 0, stream>>>(X, T, out);
}